// NadeLayer_44959717655091
// MI455X (gfx1250) — compile-verified
//
#include <hip/hip_runtime.h>

// NADE-style layer: out[k,i,l] = sum_{j<l} w[i,j]*x[k,j] + bias[i]
// B=128, N_h=512, size_in=1024.  Output 256MB f32 -> store-BW bound (~11us @23.3TB/s).
//
// Blocked segmented scan expressed as WMMA (per 16x16 output tile, i-rows x l-cols):
//   D = T_diag(16x16) x S16  +  acc (x) ones
// where T[i,j]=w[i,j]*x[k,j] on the diagonal j-block (== l-block), S16 is the
// constant strictly-upper-triangular ones matrix, and acc is the running carry
// (bias + prefix over previous blocks).  K=16 is realized as four chained
// V_WMMA_F32_16X16X4_F32 slabs; the carry rank-1 update (acc (x) ones) is a
// FIFTH chained WMMA (A = acc in K-slot 0, B = ones row) — this removes all
// ds_bpermute broadcasts from the inner loop.

typedef __attribute__((ext_vector_type(2))) float v2f;
typedef __attribute__((ext_vector_type(8))) float v8f;

#define NB   128
#define NH   512
#define SIN  1024

__global__ __launch_bounds__(256) void nade_wmma_kernel(
    const float* __restrict__ x,      // [128,1024]
    const float* __restrict__ w,      // [512,1024]
    const float* __restrict__ bias,   // [512]
    float* __restrict__ out)          // [128,512,1024]
{
    const int lane = threadIdx.x & 31;
    const int wid  = blockIdx.x * 8 + (threadIdx.x >> 5);   // 4096 waves total
    const int k    = wid >> 5;          // 32 i-blocks per batch row
    const int ib   = wid & 31;
    const int i0   = ib * 16;
    const int m    = lane & 15;         // A-matrix M index / B,C,D N(column) index
    const int hi   = lane >> 4;         // half-wave select

    const float* wrow = w + (size_t)(i0 + m) * SIN;   // this lane's weight row
    const float* xrow = x + (size_t)k * SIN;

    // running carry per i-row: bias + sum_{j < lb*16} T[m,j]
    float acc = bias[i0 + m];

    // Constant B slabs of S16 (strictly upper: S[j,n]=1 iff j<n).
    // B 4x16 f32 layout (mirrors A): VGPR v, lane L -> K = v + 2*(L>=16), N = L&15.
    v2f bq[4];
#pragma unroll
    for (int q = 0; q < 4; ++q) {
        const int j0 = 4 * q + 2 * hi;
        bq[q].x = (j0     < m) ? 1.0f : 0.0f;
        bq[q].y = (j0 + 1 < m) ? 1.0f : 0.0f;
    }
    // Carry-broadcast B: row K=0 all ones, rows K=1..3 zero.
    v2f bones;
    bones.x = (hi == 0) ? 1.0f : 0.0f;   // K = 0 (lanes 0-15) / K = 2 (lanes 16-31)
    bones.y = 0.0f;                      // K = 1 / K = 3

#pragma unroll 4
    for (int lb = 0; lb < 64; ++lb) {
        const int jb = lb * 16;

        // A 16x4 f32 layout: VGPR v, lane L -> M = L&15, K = v + 2*(L>=16).
        // Slab q covers j = jb + 4q + {0..3}; this lane needs the pair
        // j = jb + 4q + 2*hi + {0,1}  -> one float2 load each from w and x.
        v2f wv[4], xv[4];
#pragma unroll
        for (int q = 0; q < 4; ++q) {
            const int j0 = jb + 4 * q + 2 * hi;
            wv[q] = *(const v2f*)(wrow + j0);
            xv[q] = *(const v2f*)(xrow + j0);   // uniform per half-wave, L2/L0 hit
        }

        v2f  aq[4];
        float part = 0.0f;                       // this lane's share of chunk rowsum
#pragma unroll
        for (int q = 0; q < 4; ++q) {
            aq[q].x = wv[q].x * xv[q].x;
            aq[q].y = wv[q].y * xv[q].y;
            part += aq[q].x + aq[q].y;
        }

        // Carry rank-1 slab: A[m, K=0] = acc[m], everything else 0.
        v2f acar;
        acar.x = (hi == 0) ? acc : 0.0f;
        acar.y = 0.0f;

        // Chained WMMAs: start from inline-0 C, add carry outer product, then
        // the four K=4 slabs of the triangular block-matmul.
        v8f c = {};   // lowers to inline 0 C-operand on the first WMMA
        c = __builtin_amdgcn_wmma_f32_16x16x4_f32(false, acar, false, bones,
                                                  (short)0, c, false, false);
#pragma unroll
        for (int q = 0; q < 4; ++q) {
            c = __builtin_amdgcn_wmma_f32_16x16x4_f32(false, aq[q], false, bq[q],
                                                      (short)0, c, false, false);
        }

        // Store tile: D VGPR r -> out[k, i0 + r + 8*hi, jb + (lane&15)].
        // Each half-wave writes a full aligned 64B line; stream past caches (NT).
        float* orow = out + ((size_t)(k * NH + i0 + hi * 8)) * SIN + jb + m;
#pragma unroll
        for (int r = 0; r < 8; ++r) {
            __builtin_nontemporal_store(c[r], orow + (size_t)r * SIN);
        }

        // carry += full chunk rowsum (combine the two half-wave partials)
        acc += part + __shfl_xor(part, 16, 32);
    }
}

extern "C" void kernel_launch(void* const* d_in, const int* in_sizes, int n_in,
                              void* d_out, int out_size, void* d_ws, size_t ws_size,
                              hipStream_t stream) {
    const float* x    = (const float*)d_in[0];   // [128,1024]
    const float* w    = (const float*)d_in[1];   // [512,1024]
    const float* bias = (const float*)d_in[2];   // [512]
    float* out        = (float*)d_out;           // [128,512,1024]

    // 4096 waves = 512 blocks x 256 threads (8 waves/block)
    nade_wmma_kernel<<<dim3(512), dim3(256), 0, stream>>>(x, w, bias, out);
}